// FastFood_Layer_64072322122210
// MI455X (gfx1250) — compile-verified
//
#include <hip/hip_runtime.h>
#include <math.h>

typedef float v2f __attribute__((ext_vector_type(2)));
typedef float v8f __attribute__((ext_vector_type(8)));

// Within-wave LDS ordering fence: DS ops from one wave complete in order;
// the wait + memory clobber keeps both HW and compiler from reordering
// cross-lane LDS dependencies.
#define DSFENCE() asm volatile("s_wait_dscnt 0" ::: "memory")

__device__ __forceinline__ float had16(int r, int c) {
    return (__builtin_popcount(r & c) & 1) ? -1.0f : 1.0f;
}

// AXIS 0: transform along stride-64 axis (bits 9:6). addr = row*64 + col.
// AXIS 1: transform along stride-4 axis (bits 5:2).  col encodes (i,k):
//         addr = (col>>2)*64 + row*4 + (col&3).
template <int AXIS>
__device__ __forceinline__ int stage_addr(int row, int col) {
    return (AXIS == 0) ? (row * 64 + col)
                       : ((col >> 2) * 64 + row * 4 + (col & 3));
}

// One H16 stage over 1024 elements: D(16x16) = H16(16x16) * W(16x16) per
// column tile, K split into 4 chunks of 4 for V_WMMA_F32_16X16X4_F32.
// af[q] holds the A-fragment of H16 chunk q (computed once per wave).
template <int AXIS>
__device__ __forceinline__ void h16_stage(const float* src, float* dst,
                                          int l, int half, const v2f* af) {
#pragma unroll
    for (int t = 0; t < 4; ++t) {
        const int col = t * 16 + l;
        v8f acc = {0.f, 0.f, 0.f, 0.f, 0.f, 0.f, 0.f, 0.f};
#pragma unroll
        for (int q = 0; q < 4; ++q) {
            const int r0 = 4 * q + 2 * half;  // B frag: VGPR0 rows {0,2}, VGPR1 rows {1,3}
            v2f b;
            b.x = src[stage_addr<AXIS>(r0, col)];
            b.y = src[stage_addr<AXIS>(r0 + 1, col)];
            acc = __builtin_amdgcn_wmma_f32_16x16x4_f32(
                false, af[q], false, b, (short)0, acc, false, false);
        }
        // D layout: VGPR r -> M = r (lanes 0-15) or r+8 (lanes 16-31), N = l
#pragma unroll
        for (int r = 0; r < 8; ++r) {
            dst[stage_addr<AXIS>(r + 8 * half, col)] = acc[r];
        }
    }
}

// H4 over bits 1:0 — per-lane-disjoint contiguous groups of 4 (in place).
__device__ __forceinline__ void h4_stage(float* buf, int lane) {
#pragma unroll
    for (int v = 0; v < 8; ++v) {
        float4* p = (float4*)(buf + (v * 32 + lane) * 4);
        float4 d = *p;
        float a = d.x + d.y, b = d.x - d.y;
        float c = d.z + d.w, e = d.z - d.w;
        d.x = a + c; d.y = b + e; d.z = a - c; d.w = b - e;
        *p = d;
    }
}

// Full 1024-point FWHT; result ends up back in bufIn.
__device__ __forceinline__ void fwht1024(float* bufIn, float* bufTmp,
                                         int lane, int l, int half,
                                         const v2f* af) {
    h4_stage(bufIn, lane);
    DSFENCE();
    h16_stage<1>(bufIn, bufTmp, l, half, af);
    DSFENCE();
    h16_stage<0>(bufTmp, bufIn, l, half, af);
    DSFENCE();
}

__global__ __launch_bounds__(256) void fastfood_kernel(
    const float* __restrict__ xg, const float* __restrict__ Bg,
    const float* __restrict__ Gg, const float* __restrict__ Sg,
    const float* __restrict__ Ug, const long long* __restrict__ Pg,
    float* __restrict__ out, int n, int totalWaves) {
    extern __shared__ float smem[];
    float* cB  = smem;            // 4096 f32
    float* cG  = cB + 4096;       // 4096 f32
    float* cS  = cG + 4096;       // 4096 f32 (norm folded in)
    float* cPh = cS + 4096;       // 4096 f32 (2*pi*U)
    int*   cP  = (int*)(cPh + 4096);      // 4096 i32
    float* wbase = (float*)(cP + 4096);   // 8 waves * 2 * 1024 f32

    const int tid = threadIdx.x;
    const float normS = 0.03125f;  // 1/(SCALE*sqrt(1024))
    for (int i = tid; i < 4096; i += 256) {
        cB[i]  = Bg[i];
        cG[i]  = Gg[i];
        cS[i]  = Sg[i] * normS;
        cPh[i] = 6.28318530717958647692f * Ug[i];
        cP[i]  = (int)Pg[i];
    }
    __syncthreads();

    const int lane = tid & 31;
    const int wv   = tid >> 5;
    const int l    = lane & 15;
    const int half = lane >> 4;
    float* bufA = wbase + wv * 2048;
    float* bufB = bufA + 1024;

    // A fragments of H16 (ISA 16x4 A layout: VGPR0 = K 0/2, VGPR1 = K 1/3)
    v2f af[4];
#pragma unroll
    for (int q = 0; q < 4; ++q) {
        af[q].x = had16(l, 4 * q + 2 * half);
        af[q].y = had16(l, 4 * q + 2 * half + 1);
    }

    const float outScale = 0.02209708691207961101f;  // sqrt(2/4096)

    for (int s = blockIdx.x * 8 + wv; s < n; s += totalWaves) {
        const float4* x4 = (const float4*)(xg + (size_t)s * 1024);
        float* outRow = out + (size_t)s * 4096;
#pragma unroll 1
        for (int m = 0; m < 4; ++m) {
            const float* cBm  = cB  + m * 1024;
            const float* cGm  = cG  + m * 1024;
            const float* cSm  = cS  + m * 1024;
            const float* cPhm = cPh + m * 1024;
            const int*   cPm  = cP  + m * 1024;

            // bufA = x * B[m]
#pragma unroll
            for (int v = 0; v < 8; ++v) {
                const int g = v * 32 + lane;
                float4 xv = x4[g];
                float4 bv = *(const float4*)(cBm + g * 4);
                xv.x *= bv.x; xv.y *= bv.y; xv.z *= bv.z; xv.w *= bv.w;
                *(float4*)(bufA + g * 4) = xv;
            }
            DSFENCE();
            fwht1024(bufA, bufB, lane, l, half, af);  // HBx in bufA

            // bufB = HBx[P] * G
#pragma unroll
            for (int v = 0; v < 8; ++v) {
                const int g = v * 32 + lane;
                int4   pv = *(const int4*)(cPm + g * 4);
                float4 gv = *(const float4*)(cGm + g * 4);
                float4 r;
                r.x = bufA[pv.x] * gv.x;
                r.y = bufA[pv.y] * gv.y;
                r.z = bufA[pv.z] * gv.z;
                r.w = bufA[pv.w] * gv.w;
                *(float4*)(bufB + g * 4) = r;
            }
            DSFENCE();
            fwht1024(bufB, bufA, lane, l, half, af);  // HGPHBx in bufB

            // out = cos(val*S + phase) * outScale, coalesced float4 stores
#pragma unroll
            for (int v = 0; v < 8; ++v) {
                const int g = v * 32 + lane;
                float4 hv = *(const float4*)(bufB + g * 4);
                float4 sv = *(const float4*)(cSm + g * 4);
                float4 ph = *(const float4*)(cPhm + g * 4);
                float4 o;
                o.x = cosf(hv.x * sv.x + ph.x) * outScale;
                o.y = cosf(hv.y * sv.y + ph.y) * outScale;
                o.z = cosf(hv.z * sv.z + ph.z) * outScale;
                o.w = cosf(hv.w * sv.w + ph.w) * outScale;
                *(float4*)(outRow + m * 1024 + g * 4) = o;
            }
            DSFENCE();  // bufB reads complete before next m's stages rewrite it
        }
    }
}

extern "C" void kernel_launch(void* const* d_in, const int* in_sizes, int n_in,
                              void* d_out, int out_size, void* d_ws, size_t ws_size,
                              hipStream_t stream) {
    (void)n_in; (void)d_ws; (void)ws_size; (void)out_size;
    const float*     x = (const float*)d_in[0];
    const float*     B = (const float*)d_in[1];
    const float*     G = (const float*)d_in[2];
    const float*     S = (const float*)d_in[3];
    const float*     U = (const float*)d_in[4];
    const long long* P = (const long long*)d_in[5];

    const int n = in_sizes[0] / 1024;
    if (n <= 0) return;

    int blocks = (n + 7) / 8;
    if (blocks > 512) blocks = 512;
    const int totalWaves = blocks * 8;

    // LDS: 5*16KB constants + 8 waves * 8KB ping-pong = 144KB
    const size_t shmem = (size_t)(4 * 4096 + 4096 + 8 * 2 * 1024) * sizeof(float);

    fastfood_kernel<<<blocks, 256, shmem, stream>>>(x, B, G, S, U, P,
                                                    (float*)d_out, n, totalWaves);
}